// VectorQuantizer_16741782520497
// MI455X (gfx1250) — compile-verified
//
#include <hip/hip_runtime.h>
#include <stdint.h>

// ---------------------------------------------------------------------------
// VQ codebook lookup for MI455X (gfx1250, wave32).
//   distances via V_WMMA_F32_16X16X4_F32 (exact fp32 -> safe argmin)
//   codebook tiles double-buffered in LDS via GLOBAL_LOAD_ASYNC_TO_LDS_B128
//   (ASYNCcnt-tracked), bank-conflict-free padded layout, pipelined ds reads
// ---------------------------------------------------------------------------

typedef __attribute__((ext_vector_type(2))) float v2f;
typedef __attribute__((ext_vector_type(8))) float v8f;

#define VQ_BETA   0.25f
#define D_MODEL   128
#define K_CODES   1024
#define T_TOKENS  65536            // 8*64*128 tokens
#define N_TILES   (K_CODES / 16)   // 64 codebook tiles of 16 codes
#define K_STEPS   (D_MODEL / 4)    // 32 wmma k-steps (K=4 each)
#define TILES_M   (T_TOKENS / 16)  // 4096 token tiles
#define ROW_PAD   132              // LDS row stride in floats (132 % 64 == 4)

// ---- codebook squared norms: one wave per code -----------------------------
__global__ __launch_bounds__(128) void vq_cnorm_kernel(
    const float* __restrict__ cb, float* __restrict__ cnorm) {
  const int wave = threadIdx.x >> 5;
  const int lane = threadIdx.x & 31;
  const int code = blockIdx.x * 4 + wave;
  const float4 v = *(reinterpret_cast<const float4*>(cb + (size_t)code * D_MODEL) + lane);
  float s = v.x * v.x + v.y * v.y + v.z * v.z + v.w * v.w;
  for (int off = 16; off; off >>= 1) s += __shfl_xor(s, off, 32);
  if (lane == 0) cnorm[code] = s;
}

// ---- main kernel: 4 waves/block, 1 wave = 16 tokens vs all 1024 codes ------
__global__ __launch_bounds__(128) void vq_main_kernel(
    const float* __restrict__ x, const float* __restrict__ cb,
    const float* __restrict__ cnorm, float* __restrict__ xq,
    float* __restrict__ lpart) {
  // Double-buffered codebook tile: 16 rows x 128 floats, padded to 132 so that
  // a B-fragment ds_load_b64 (lane i, half h -> banks {4i+2h, 4i+2h+1}) is
  // conflict-free across all 32 lanes.
  __shared__ float smem[2][16 * ROW_PAD];
  __shared__ int   idx_s[4][16];

  const int tid  = threadIdx.x;
  const int wave = tid >> 5;
  const int lane = tid & 31;
  const int l15  = lane & 15;
  const int half = lane >> 4;          // 0: K%4 in {0,1}, 1: K%4 in {2,3}
  const int tile = blockIdx.x * 4 + wave;
  const int tok0 = tile * 16;

  // ---- A fragments: row (tok0+l15), float2 at K%4 = 2*half, kept in regs ---
  const float* xr = x + (size_t)(tok0 + l15) * D_MODEL + half * 2;
  v2f a[K_STEPS];
#pragma unroll
  for (int j = 0; j < K_STEPS; ++j)
    a[j] = *reinterpret_cast<const v2f*>(xr + j * 4);

  // ---- async stage helpers (contiguous 8KB tile -> padded LDS) -------------
  // thread tid copies float4 chunks q = tid + 128*k : row = q>>5, col = (q&31)*4
  const int r0 = tid >> 5;             // 0..3
  const int c0 = (tid & 31) * 4;       // 0,4,...,124
  // Generic LDS pointers carry the LDS byte offset in their low 32 bits.
  const uint32_t lds0 = (uint32_t)(uintptr_t)(&smem[0][0]);
  const uint32_t ldsrow0 = lds0 + (uint32_t)(r0 * ROW_PAD + c0) * 4u;

#define STAGE_TILE(NT, BUF)                                                     \
  do {                                                                          \
    const uint64_t g0 = (uint64_t)(uintptr_t)(cb + (size_t)(NT)*2048 +          \
                                              (size_t)r0 * 128 + c0);           \
    const uint32_t l0 = ldsrow0 + (uint32_t)(BUF) * (16u * ROW_PAD * 4u);       \
    asm volatile("global_load_async_to_lds_b128 %0, %1, off"                    \
                 :: "v"(l0), "v"(g0) : "memory");                               \
    asm volatile("global_load_async_to_lds_b128 %0, %1, off"                    \
                 :: "v"(l0 + 4u * ROW_PAD * 4u), "v"(g0 + 2048ull) : "memory"); \
    asm volatile("global_load_async_to_lds_b128 %0, %1, off"                    \
                 :: "v"(l0 + 8u * ROW_PAD * 4u), "v"(g0 + 4096ull) : "memory"); \
    asm volatile("global_load_async_to_lds_b128 %0, %1, off"                    \
                 :: "v"(l0 + 12u * ROW_PAD * 4u), "v"(g0 + 6144ull) : "memory");\
  } while (0)

  float best[8];
  int   bidx[8];
#pragma unroll
  for (int v = 0; v < 8; ++v) { best[v] = 3.4e38f; bidx[v] = 0; }

  STAGE_TILE(0, 0);  // prologue: tile 0 -> buffer 0

#pragma unroll 1
  for (int nt = 0; nt < N_TILES; ++nt) {
    const int cur = nt & 1;
    // Stage next tile into the other buffer (freed by last iteration's tail
    // barrier), then wait until this tile's 4 async loads (issued last
    // iteration) are complete: async loads finish in order, so cnt<=4 works.
    if (nt + 1 < N_TILES) {
      STAGE_TILE(nt + 1, cur ^ 1);
      asm volatile("s_wait_asynccnt 4" ::: "memory");
    } else {
      asm volatile("s_wait_asynccnt 0" ::: "memory");
    }
    __syncthreads();  // all waves' portion of tile nt visible in LDS

    // ---- 16x16 distance tile: 32 x V_WMMA_F32_16X16X4_F32 -----------------
    // B fragment: lane holds code row l15, float2 at K%4 = 2*half, from LDS.
    const float* bsrc = &smem[cur][l15 * ROW_PAD + half * 2];
    v8f acc = {0.f, 0.f, 0.f, 0.f, 0.f, 0.f, 0.f, 0.f};
    v2f bb[2][8];
#pragma unroll
    for (int i = 0; i < 8; ++i)
      bb[0][i] = *reinterpret_cast<const v2f*>(bsrc + i * 4);
#pragma unroll
    for (int g = 0; g < 4; ++g) {
      if (g < 3) {
#pragma unroll
        for (int i = 0; i < 8; ++i)
          bb[(g + 1) & 1][i] =
              *reinterpret_cast<const v2f*>(bsrc + (g + 1) * 32 + i * 4);
      }
#pragma unroll
      for (int i = 0; i < 8; ++i)
        acc = __builtin_amdgcn_wmma_f32_16x16x4_f32(
            false, a[g * 8 + i], false, bb[g & 1][i], (short)0, acc,
            false, false);
    }

    // score = ||e||^2 - 2 x.e (||x||^2 constant per row: argmin-invariant)
    const float cn = cnorm[nt * 16 + l15];
    const int   ci = nt * 16 + l15;
#pragma unroll
    for (int v = 0; v < 8; ++v) {
      const float s = cn - 2.0f * acc[v];
      if (s < best[v]) { best[v] = s; bidx[v] = ci; }  // strict < keeps lowest nt
    }
    __syncthreads();  // done reading smem[cur]; it is re-staged next iteration
  }

  // ---- cross-lane argmin: C reg v, lane l -> row M = v + 8*(l>>4) ----------
#pragma unroll
  for (int v = 0; v < 8; ++v) {
    float d = best[v];
    int   i = bidx[v];
    for (int off = 8; off; off >>= 1) {
      const float od = __shfl_xor(d, off, 16);
      const int   oi = __shfl_xor(i, off, 16);
      if (od < d || (od == d && oi < i)) { d = od; i = oi; }  // jnp.argmin tie
    }
    best[v] = d; bidx[v] = i;
  }
  if (l15 == 0) {
#pragma unroll
    for (int v = 0; v < 8; ++v) idx_s[wave][v + 8 * half] = bidx[v];
  }
  __syncthreads();

  // ---- gather winners: coalesced float4 codebook row -> x_q, loss partial --
  const float* xbase = x + (size_t)tok0 * D_MODEL;
  float ls = 0.f;
#pragma unroll 1
  for (int r = 0; r < 16; ++r) {
    const int id = idx_s[wave][r];
    const float4 c4 = *(reinterpret_cast<const float4*>(cb + (size_t)id * D_MODEL) + lane);
    const float4 x4 = *(reinterpret_cast<const float4*>(xbase + (size_t)r * D_MODEL) + lane);
    *(reinterpret_cast<float4*>(xq + (size_t)(tok0 + r) * D_MODEL) + lane) = c4;
    const float dx = c4.x - x4.x, dy = c4.y - x4.y;
    const float dz = c4.z - x4.z, dw = c4.w - x4.w;
    ls += dx * dx + dy * dy + dz * dz + dw * dw;
  }
  for (int off = 16; off; off >>= 1) ls += __shfl_xor(ls, off, 32);
  if (lane == 0) lpart[tile] = ls;
#undef STAGE_TILE
}

// ---- deterministic loss reduction: 8 tiles per (b,s), fixed order ----------
__global__ void vq_loss_kernel(const float* __restrict__ lpart,
                               float* __restrict__ loss) {
  const int bs = blockIdx.x * blockDim.x + threadIdx.x;
  if (bs < 512) {
    float s = 0.f;
#pragma unroll
    for (int t = 0; t < 8; ++t) s += lpart[bs * 8 + t];
    loss[bs] = s * ((1.0f + VQ_BETA) / (float)(128 * 128));
  }
}

extern "C" void kernel_launch(void* const* d_in, const int* in_sizes, int n_in,
                              void* d_out, int out_size, void* d_ws, size_t ws_size,
                              hipStream_t stream) {
  const float* x  = (const float*)d_in[0];   // [8,64,128,128] fp32
  const float* cb = (const float*)d_in[1];   // [1024,128] fp32

  float* out  = (float*)d_out;
  float* xq   = out;                               // 8,388,608 floats
  float* loss = out + (size_t)T_TOKENS * D_MODEL;  // 512 floats

  float* cnorm = (float*)d_ws;               // 1024 floats
  float* lpart = (float*)d_ws + K_CODES;     // 4096 floats

  vq_cnorm_kernel<<<K_CODES / 4, 128, 0, stream>>>(cb, cnorm);
  vq_main_kernel<<<TILES_M / 4, 128, 0, stream>>>(x, cb, cnorm, xq, lpart);
  vq_loss_kernel<<<2, 256, 0, stream>>>(lpart, loss);
}